// ProbMask_34462817583503
// MI455X (gfx1250) — compile-verified
//
#include <hip/hip_runtime.h>
#include <hip/hip_bf16.h>

// ProbMask: out[b,h,u,col] = (col > m_top[b,h,u]) ? 1.0f : 0.0f
// B,H,U,L_K = 16,8,256,4096 -> 134.2M floats = 536 MB of pure streaming stores.
// Store-bandwidth bound (~23 us at 23.3 TB/s on MI455X). `scores` is never read.
//
// Design:
//  * one block per (b,h,u) row -> m_top[row] is block-uniform: s_load + SALU only
//  * threshold rewrite: (c + j > m) <=> (t*4 > m - j - k*1024), so every lane
//    compares ONE register (t*4) against precomputed uniform scalar thresholds;
//    zero per-iteration vector address/index arithmetic
//  * fully unrolled 4x global_store_b128 th:TH_NT from one base address with
//    immediate instruction offsets; each wave32 store covers 512 contiguous B
//  * NT hint: 536 MB write-once stream > 192 MB L2 -> stream past caches

typedef float v4f __attribute__((ext_vector_type(4)));

__device__ __forceinline__ float step_gt(int c, int thresh) {
    return (c > thresh) ? 1.0f : 0.0f;
}

__global__ __launch_bounds__(256) void
probmask_stream4k_kernel(const int* __restrict__ m_top,
                         float* __restrict__ out) {
    const int row = blockIdx.x;
    const int m = m_top[row];                    // uniform -> s_load_b32

    const int t4 = ((int)threadIdx.x) << 2;      // this lane's base column
    float* __restrict__ p = out + (size_t)row * 4096 + t4;

#pragma unroll
    for (int k = 0; k < 4; ++k) {
        const int base = k * 1024;               // compile-time constant
        v4f v;
        // All four compares are (t4 > uniform_scalar): 1 VGPR, 4 SGPR thresholds.
        v.x = step_gt(t4, m - 0 - base);
        v.y = step_gt(t4, m - 1 - base);
        v.z = step_gt(t4, m - 2 - base);
        v.w = step_gt(t4, m - 3 - base);
        // global_store_b128 th:TH_NT, immediate offset k*4096 bytes.
        __builtin_nontemporal_store(v, (v4f*)(p + base));
    }
}

// Generic fallback for L_K != 4096 (multiple of 4 assumed, as any float4 path).
__global__ __launch_bounds__(256) void
probmask_stream_kernel(const int* __restrict__ m_top,
                       float* __restrict__ out,
                       int L_K) {
    const int row = blockIdx.x;
    const int m = m_top[row];

    float* __restrict__ row_out = out + (size_t)row * (size_t)L_K;
    const int t4 = ((int)threadIdx.x) << 2;

    for (int c0 = t4; c0 + 3 < L_K; c0 += 256 * 4) {
        v4f v;
        v.x = step_gt(c0, m - 0);
        v.y = step_gt(c0, m - 1);
        v.z = step_gt(c0, m - 2);
        v.w = step_gt(c0, m - 3);
        __builtin_nontemporal_store(v, (v4f*)(row_out + c0));
    }
}

extern "C" void kernel_launch(void* const* d_in, const int* in_sizes, int n_in,
                              void* d_out, int out_size, void* d_ws, size_t ws_size,
                              hipStream_t stream) {
    // d_in[0] = scores (float32) -- unused by the reference math (shape only).
    // d_in[1] = m_top  (int32), flat count = B*H*U rows.
    const int* m_top = (const int*)d_in[1];
    float* out = (float*)d_out;

    const int n_rows = in_sizes[1];          // B*H*U = 32768
    const int L_K = out_size / n_rows;       // 4096

    dim3 grid((unsigned)n_rows);
    dim3 block(256);
    if (L_K == 4096) {
        probmask_stream4k_kernel<<<grid, block, 0, stream>>>(m_top, out);
    } else {
        probmask_stream_kernel<<<grid, block, 0, stream>>>(m_top, out, L_K);
    }
}